// ForwardWarp_25761213841994
// MI455X (gfx1250) — compile-verified
//
#include <hip/hip_runtime.h>

// ForwardWarp / CenterNet gaussian scatter-max heatmap for MI455X (gfx1250).
//
// Design (compile-only, reasoned from MI455X specs):
//  * Total data movement ~40-60MB -> ~2-3us at 23.3 TB/s; heatmap (21MB) lives
//    in the 192MB L2, so the scatter-max atomics are L2-resident. The kernel is
//    bound by atomic issue + init bandwidth, not FLOPs.
//  * One wave32 per object: the 17x17 gaussian window is computed fully
//    lane-parallel; 4096 blocks x 8 waves = 32768 objects.
//  * Gaussian is separable: g[dy,dx] = e[dy]*e[dx], e[j]=exp(-j^2/(2s^2)).
//    A rank-1 outer product == one V_WMMA_F32_16X16X4_F32 with e in column 0
//    of A and row 0 of B (lanes 0-15, first VGPR), zeros elsewhere. One WMMA
//    yields the whole 16x16 tile (offsets -8..+7) in fp32.
//  * Scatter-max: heatmap is zero-init and all gaussian values are > 0, so the
//    IEEE bit pattern is monotonic -> native GLOBAL_ATOMIC_MAX_U32, no return
//    (STOREcnt only), no CAS loop.
//  * The dy=+8 / dx=+8 rim (33 cells) only matters when r==8 (impossible for
//    these inputs, radius <= ~5.5) -> predicated VALU fallback for generality.

typedef __attribute__((ext_vector_type(2))) float v2f;
typedef __attribute__((ext_vector_type(8))) float v8f;

#define BB   128
#define NN   256
#define HH   272
#define WW   152
#define HWSZ (HH * WW)
#define RMX  8

__global__ __launch_bounds__(256) void fw_zero_kernel(float4* __restrict__ out, int n4) {
    int i = blockIdx.x * blockDim.x + threadIdx.x;
    const int stride = gridDim.x * blockDim.x;
    const float4 z = make_float4(0.f, 0.f, 0.f, 0.f);
    for (; i < n4; i += stride) out[i] = z;
}

__global__ __launch_bounds__(256) void fw_splat_kernel(
    const float* __restrict__ flow,   // [B,2,H,W]
    const float* __restrict__ wh,     // [B,N,4]
    const int*   __restrict__ mask,   // [B,N]
    const int*   __restrict__ index,  // [B,N]
    unsigned int* __restrict__ hm)    // [B,H,W] as uint bit patterns (>=0 floats)
{
    const int lane = threadIdx.x & 31;
    const int wave = threadIdx.x >> 5;
    const int obj  = blockIdx.x * 8 + wave;      // grid sized exactly B*N/8
    const int b    = obj >> 8;                   // N == 256

    // ---- per-object scalars (wave-uniform; every lane computes the same) ----
    const int   idx = index[obj];
    const float m   = (float)mask[obj];
    const float x   = flow[(b * 2 + 0) * HWSZ + idx] * m;
    const float y   = flow[(b * 2 + 1) * HWSZ + idx] * m;
    const float4 w4 = ((const float4*)wh)[obj];
    const float w   = (w4.x + w4.z) * m;
    const float h   = (w4.y + w4.w) * m;

    const bool valid = (h > 0.f) & (w > 0.f) & (x > 0.f) & (y > 0.f) &
                       (x < 152.f) & (y < 272.f);
    // uniform condition -> scalar branch, keeps EXEC all-ones at the WMMA
    if (!__builtin_amdgcn_readfirstlane((int)valid)) return;

    // ---- CenterNet gaussian radius (faithful to reference formula) ----
    const float hc = ceilf(h), wc = ceilf(w);
    const float b1 = hc + wc;
    const float c1 = wc * hc * (1.f - 0.7f) / (1.f + 0.7f);
    const float r1 = (b1 + sqrtf(b1 * b1 - 4.f * c1)) * 0.5f;
    const float b2 = 2.f * (hc + wc);
    const float c2 = (1.f - 0.7f) * wc * hc;
    const float r2 = (b2 + sqrtf(b2 * b2 - 16.f * c2)) * 0.5f;
    const float b3 = -2.f * 0.7f * (hc + wc);
    const float c3 = (0.7f - 1.f) * wc * hc;
    const float r3 = (b3 + sqrtf(b3 * b3 - 16.f * 0.7f * c3)) * 0.5f;
    const float radius = fmaxf(fminf(fminf(r1, r2), r3), 0.f);
    const float r_int  = floorf(radius);
    const int   ri     = (int)fminf(r_int, (float)RMX);   // grid statically bounded
    const float sigma  = (2.f * r_int + 1.f) * (1.f / 6.f);
    const float inv2s2 = 1.f / (2.f * sigma * sigma);

    // ---- separable gaussian vector e[j] = exp(-j^2 / (2 sigma^2)) ----
    const int   j  = lane - RMX;                 // lanes 0..15 -> j = -8..7
    const float ev = __expf(-(float)(j * j) * inv2s2);

    // A (16x4 f32, 2 VGPRs): lane<16 VGPR0 holds column K=0 -> e; rest zero.
    // B (4x16 f32, 2 VGPRs): lane<16 VGPR0 holds row    K=0 -> e; rest zero.
    // Only the K=0 column/row are nonzero => D = e (outer) e regardless of
    // where K=1..3 live in the layout.
    v2f a;
    a.x = (lane < 16) ? ev : 0.f;
    a.y = 0.f;
    v8f c = {};
    v8f d = __builtin_amdgcn_wmma_f32_16x16x4_f32(
        /*neg_a=*/false, a, /*neg_b=*/false, a,
        /*c_mod=*/(short)0, c, /*reuse_a=*/false, /*reuse_b=*/false);

    // ---- scatter-max the 16x16 tile: D vgpr v, lanes 0-15 -> (M=v, N=lane),
    //      lanes 16-31 -> (M=v+8, N=lane-16); dy = M-8, dx = N-8 ----
    const int yi = (int)floorf(y);
    const int xi = (int)floorf(x);
    unsigned int* __restrict__ hb = hm + b * HWSZ;

    const int dx    = (lane & 15) - RMX;
    const int xx    = xi + dx;
    const int mbase = (lane >> 4) * 8;
    const bool xok  = (dx >= -ri) & (dx <= ri) & (xx >= 0) & (xx < WW);

#pragma unroll
    for (int v = 0; v < 8; ++v) {
        const int dy = mbase + v - RMX;
        const int yy = yi + dy;
        const float g = d[v];
        const bool ok = xok & (dy >= -ri) & (dy <= ri) &
                        (yy >= 0) & (yy < HH) & (g >= 2e-15f);
        if (ok) atomicMax(hb + yy * WW + xx, __float_as_uint(g));
    }

    // ---- rim cells dy=+8 / dx=+8: only reachable when r_int >= 8 ----
    if (ri >= RMX) {
        {   // dy = +8 row, dx = lane-8 for lanes 0..16
            const int dye = RMX;
            const int dxe = lane - RMX;
            const int yy = yi + dye, xx2 = xi + dxe;
            const float g = __expf(-(float)(dxe * dxe + dye * dye) * inv2s2);
            const bool ok = (dxe <= RMX) & (yy >= 0) & (yy < HH) &
                            (xx2 >= 0) & (xx2 < WW) & (g >= 2e-15f);
            if (ok) atomicMax(hb + yy * WW + xx2, __float_as_uint(g));
        }
        {   // dx = +8 column, dy = lane-8 for lanes 0..15 (corner already done)
            const int dxe = RMX;
            const int dye = lane - RMX;
            const int yy = yi + dye, xx2 = xi + dxe;
            const float g = __expf(-(float)(dxe * dxe + dye * dye) * inv2s2);
            const bool ok = (dye <= RMX - 1) & (yy >= 0) & (yy < HH) &
                            (xx2 >= 0) & (xx2 < WW) & (g >= 2e-15f);
            if (ok) atomicMax(hb + yy * WW + xx2, __float_as_uint(g));
        }
    }
}

extern "C" void kernel_launch(void* const* d_in, const int* in_sizes, int n_in,
                              void* d_out, int out_size, void* d_ws, size_t ws_size,
                              hipStream_t stream) {
    const float* flow  = (const float*)d_in[0];
    const float* wh    = (const float*)d_in[1];
    const int*   mask  = (const int*)d_in[2];
    const int*   index = (const int*)d_in[3];
    float* out = (float*)d_out;

    // out_size = 128*272*152 = 5,292,032 (divisible by 4)
    const int n4 = out_size >> 2;
    fw_zero_kernel<<<1024, 256, 0, stream>>>((float4*)out, n4);

    // one wave32 per object; 8 waves per 256-thread block
    fw_splat_kernel<<<(BB * NN) / 8, 256, 0, stream>>>(
        flow, wh, mask, index, (unsigned int*)out);
}